// VectorQuantizer_27487790694441
// MI455X (gfx1250) — compile-verified
//
#include <hip/hip_runtime.h>

typedef __attribute__((ext_vector_type(2))) float v2f;
typedef __attribute__((ext_vector_type(8))) float v8f;

#define D               64
#define K_CODES         1024
#define N_ROWS          65536
#define ROWS_PER_WAVE   16
#define WAVES_PER_BLOCK 8
#define ROWS_PER_BLOCK  (ROWS_PER_WAVE * WAVES_PER_BLOCK)   // 128
#define N_BLOCKS        (N_ROWS / ROWS_PER_BLOCK)           // 512
#define CODES_PER_STAGE 128
#define N_STAGES        (K_CODES / CODES_PER_STAGE)         // 8
#define TILES_PER_STAGE (CODES_PER_STAGE / 16)              // 8
#define STAGE_FLOATS    (CODES_PER_STAGE * D)               // 8192 (32 KB)

// d_out layout (reference returns (quantized_st, loss, indices[:,None]) concatenated):
//   [0, 4194304)            quantized (f32)
//   [4194304]               loss (f32)
//   [4194305, 4259841)      encoding indices (as f32)
#define Q_ELEMS   (N_ROWS * D)

__global__ __launch_bounds__(256)
void vq_main_kernel(const float* __restrict__ X, const float* __restrict__ W,
                    float* __restrict__ outQ, float* __restrict__ outIdx,
                    float* __restrict__ wsPartial)
{
    // Double-buffered codebook staging: 2 x 32 KB + 8 floats of loss scratch.
    extern __shared__ float smem[];
    float* ldsLoss = smem + 2 * STAGE_FLOATS;

    const int tid  = threadIdx.x;
    const int lane = tid & 31;
    const int wave = tid >> 5;
    const int hi   = lane >> 4;      // which half of the wave
    const int lm   = lane & 15;      // lane within half-group
    const int rowbase = blockIdx.x * ROWS_PER_BLOCK + wave * ROWS_PER_WAVE;

    // ---- Load A-fragments: X tile 16x64, resident in VGPRs for the whole kernel.
    // ISA A layout (16x4 f32): lanes 0-15 -> M=lane, K={4t,4t+1}; lanes 16-31 -> K={4t+2,4t+3}.
    v2f a[16];
    {
        const float* xrow = X + (size_t)(rowbase + lm) * D + 2 * hi;
        #pragma unroll
        for (int t = 0; t < 16; ++t)
            a[t] = *(const v2f*)(xrow + 4 * t);
    }

    float bestKey[8];
    int   bestIdx[8];
    #pragma unroll
    for (int v = 0; v < 8; ++v) { bestKey[v] = 3.4e38f; bestIdx[v] = 0; }

    // ---- Prologue: stage chunk 0 into buffer 0.
    {
        const float4* src = (const float4*)W;
        float4* dst = (float4*)smem;
        float4 r[8];
        #pragma unroll
        for (int k = 0; k < 8; ++k) r[k] = src[tid + k * 256];
        #pragma unroll
        for (int k = 0; k < 8; ++k) dst[tid + k * 256] = r[k];
    }
    __syncthreads();

    for (int stage = 0; stage < N_STAGES; ++stage) {
        const float* curW = smem + (stage & 1) * STAGE_FLOATS;

        // Issue next chunk's global loads early; they complete under the WMMA work.
        float4 r[8];
        const bool haveNext = (stage + 1) < N_STAGES;
        if (haveNext) {
            const float4* src = (const float4*)(W + (size_t)(stage + 1) * STAGE_FLOATS);
            #pragma unroll
            for (int k = 0; k < 8; ++k) r[k] = src[tid + k * 256];
        }

        for (int ct = 0; ct < TILES_PER_STAGE; ++ct) {
            // B layout (4x16 f32, K x N): lane holds code n=lm, cols {4t+2hi, 4t+2hi+1}.
            const float* wrow = curW + (ct * 16 + lm) * D + 2 * hi;

            // Batch all B-fragments first so LDS latency is paid once per tile,
            // then stream the WMMA chain back-to-back.
            v2f b[16];
            #pragma unroll
            for (int t = 0; t < 16; ++t)
                b[t] = *(const v2f*)(wrow + 4 * t);

            float wpart = 0.f;
            #pragma unroll
            for (int t = 0; t < 16; ++t) {
                wpart = fmaf(b[t].x, b[t].x, wpart);
                wpart = fmaf(b[t].y, b[t].y, wpart);
            }

            // Two independent accumulator chains to halve WMMA RAW depth.
            v8f acc0 = {};
            v8f acc1 = {};
            #pragma unroll
            for (int t = 0; t < 16; t += 2) {
                acc0 = __builtin_amdgcn_wmma_f32_16x16x4_f32(
                         false, a[t],     false, b[t],     (short)0, acc0, false, false);
                acc1 = __builtin_amdgcn_wmma_f32_16x16x4_f32(
                         false, a[t + 1], false, b[t + 1], (short)0, acc1, false, false);
            }

            // lane L and L^16 cover complementary halves of the same code row
            float wnorm = wpart + __shfl_xor(wpart, 16);
            const int code = stage * CODES_PER_STAGE + ct * 16 + lm;
            // argmin key: ||w||^2 - 2 x.w   (||x||^2 constant per row -> dropped)
            #pragma unroll
            for (int v = 0; v < 8; ++v) {
                float dot = acc0[v] + acc1[v];
                float key = fmaf(-2.f, dot, wnorm);
                if (key < bestKey[v]) { bestKey[v] = key; bestIdx[v] = code; }
            }
        }

        if (haveNext) {
            float4* dst = (float4*)(smem + ((stage + 1) & 1) * STAGE_FLOATS);
            #pragma unroll
            for (int k = 0; k < 8; ++k) dst[tid + k * 256] = r[k];
            __syncthreads();
        }
    }

    // ---- Reduce argmin across the 16 lanes of each half-group (rows v+8*hi).
    #pragma unroll
    for (int s = 1; s <= 8; s <<= 1) {
        #pragma unroll
        for (int v = 0; v < 8; ++v) {
            float ok = __shfl_xor(bestKey[v], s);
            int   oi = __shfl_xor(bestIdx[v], s);
            if (ok < bestKey[v] || (ok == bestKey[v] && oi < bestIdx[v])) {
                bestKey[v] = ok; bestIdx[v] = oi;
            }
        }
    }

    // ---- Epilogue: gather W[best] (L2-hot), write quantized + indices, accumulate MSE.
    float sumsq = 0.f;
    #pragma unroll
    for (int m = 0; m < 16; ++m) {
        // rows 0-7 live in lanes 0-15 (v=m), rows 8-15 in lanes 16-31 (v=m-8)
        int bidx = __shfl(bestIdx[m & 7], (m >> 3) << 4);
        int row  = rowbase + m;
        const float* qp = W + (size_t)bidx * D + 2 * lane;
        const float* xp = X + (size_t)row  * D + 2 * lane;
        float q0 = qp[0], q1 = qp[1];
        float x0 = xp[0], x1 = xp[1];
        float* op = outQ + (size_t)row * D + 2 * lane;
        op[0] = q0; op[1] = q1;
        float d0 = q0 - x0, d1 = q1 - x1;
        sumsq = fmaf(d0, d0, sumsq);
        sumsq = fmaf(d1, d1, sumsq);
        if (lane == 0) outIdx[row] = (float)bidx;
    }
    // deterministic wave-tree reduction
    #pragma unroll
    for (int s = 16; s >= 1; s >>= 1) sumsq += __shfl_xor(sumsq, s);
    if (lane == 0) ldsLoss[wave] = sumsq;
    __syncthreads();
    if (tid == 0) {
        float t = 0.f;
        #pragma unroll
        for (int w = 0; w < WAVES_PER_BLOCK; ++w) t += ldsLoss[w];
        wsPartial[blockIdx.x] = t;
    }
}

// Fixed-order final reduction of 512 block partials -> deterministic loss.
__global__ __launch_bounds__(32)
void vq_loss_kernel(const float* __restrict__ wsPartial, float* __restrict__ lossOut)
{
    int lane = threadIdx.x;
    float s = 0.f;
    #pragma unroll
    for (int k = 0; k < N_BLOCKS / 32; ++k)
        s += wsPartial[lane + 32 * k];
    #pragma unroll
    for (int m = 16; m >= 1; m >>= 1) s += __shfl_xor(s, m);
    if (lane == 0)
        *lossOut = s * (1.25f / (float)Q_ELEMS);   // q_loss + 0.25*e_loss = 1.25*MSE
}

extern "C" void kernel_launch(void* const* d_in, const int* in_sizes, int n_in,
                              void* d_out, int out_size, void* d_ws, size_t ws_size,
                              hipStream_t stream)
{
    const float* X = (const float*)d_in[0];   // (16,64,64,64) f32 -> 65536 x 64
    const float* W = (const float*)d_in[1];   // (1024, 64) f32
    float* out    = (float*)d_out;
    float* outQ   = out;
    float* loss   = out + Q_ELEMS;
    float* outIdx = out + Q_ELEMS + 1;
    float* ws     = (float*)d_ws;             // 512 block partials

    size_t shmem = (2 * STAGE_FLOATS + WAVES_PER_BLOCK) * sizeof(float);  // ~64 KB
    vq_main_kernel<<<N_BLOCKS, 256, shmem, stream>>>(X, W, outQ, outIdx, ws);
    vq_loss_kernel<<<1, 32, 0, stream>>>(ws, loss);
}